// MolecularGNN_29506425324199
// MI455X (gfx1250) — compile-verified
//
#include <hip/hip_runtime.h>
#include <math.h>

typedef __attribute__((ext_vector_type(2))) float v2f;
typedef __attribute__((ext_vector_type(8))) float v8f;

#define N_NODES   50000
#define N_EDGES   600000
#define ATOM_DIM  75
#define HID       128
#define N_LAYERS  4
#define NUM_GRAPHS 128
#define EPSBN     1e-5f

__device__ __forceinline__ v2f ld2(const float* __restrict__ p)
{
    v2f r; r.x = p[0]; r.y = p[1]; return r;
}

struct Frag {
    v2f a, b0, b1, b2, b3;
};

__device__ __forceinline__ Frag load_frag(const float* __restrict__ arow,
                                          const float* __restrict__ br0,
                                          const float* __restrict__ br1,
                                          const float* __restrict__ br2,
                                          const float* __restrict__ br3, int ka)
{
    Frag f;
    f.a  = ld2(arow + ka);
    f.b0 = ld2(br0 + ka);
    f.b1 = ld2(br1 + ka);
    f.b2 = ld2(br2 + ka);
    f.b3 = ld2(br3 + ka);
    return f;
}

__device__ __forceinline__ void wmma_step(const Frag& f, v8f acc[4])
{
    acc[0] = __builtin_amdgcn_wmma_f32_16x16x4_f32(false, f.a, false, f.b0, (short)0, acc[0], false, false);
    acc[1] = __builtin_amdgcn_wmma_f32_16x16x4_f32(false, f.a, false, f.b1, (short)0, acc[1], false, false);
    acc[2] = __builtin_amdgcn_wmma_f32_16x16x4_f32(false, f.a, false, f.b2, (short)0, acc[2], false, false);
    acc[3] = __builtin_amdgcn_wmma_f32_16x16x4_f32(false, f.a, false, f.b3, (short)0, acc[3], false, false);
}

// Accumulate A[m0:m0+16, :] @ W[n0:n0+64, :]^T into acc[4] (16x64 tile).
// Depth-2 software pipeline (3 fragment sets in flight), #pragma unroll 1 so
// the compiler cannot re-serialize loads against the WMMAs; a single guarded
// tail iteration handles K % 4 != 0 (only the K=75 embedding GEMM).
__device__ __forceinline__ void gemm_accum(const float* __restrict__ A, const float* __restrict__ W,
                                           int K, int m0, int n0, int l, int hh, v8f acc[4])
{
    const float* arow = A + (size_t)(m0 + l) * K;
    const float* br0  = W + (size_t)(n0 + l) * K;
    const float* br1  = br0 + (size_t)16 * K;
    const float* br2  = br0 + (size_t)32 * K;
    const float* br3  = br0 + (size_t)48 * K;
    const int Ka = K & ~3;
    const int kb = 2 * hh;

    if (Ka >= 8) {
        Frag f0 = load_frag(arow, br0, br1, br2, br3, kb);
        Frag f1 = load_frag(arow, br0, br1, br2, br3, kb + 4);
        #pragma unroll 1
        for (int k0 = 8; k0 < Ka; k0 += 4) {
            Frag fn = load_frag(arow, br0, br1, br2, br3, kb + k0);
            wmma_step(f0, acc);
            f0 = f1;
            f1 = fn;
        }
        wmma_step(f0, acc);
        wmma_step(f1, acc);
    } else if (Ka > 0) {
        Frag f0 = load_frag(arow, br0, br1, br2, br3, kb);
        wmma_step(f0, acc);
    }
    if (Ka < K) {               // uniform branch; taken only when K % 4 != 0
        const int kt = Ka + kb;
        Frag f = {};
        if (kt < K) {
            f.a.x = arow[kt]; f.b0.x = br0[kt]; f.b1.x = br1[kt];
            f.b2.x = br2[kt]; f.b3.x = br3[kt];
        }
        if (kt + 1 < K) {
            f.a.y = arow[kt + 1]; f.b0.y = br0[kt + 1]; f.b1.y = br1[kt + 1];
            f.b2.y = br2[kt + 1]; f.b3.y = br3[kt + 1];
        }
        wmma_step(f, acc);
    }
}

// ---------------------------------------------------------------------------
// WMMA f32 GEMM:  C[M,Nout] = act( A1 @ W1^T (+ A2 @ W2^T) + b1 (+ b2) )
// One wave per block; each wave computes a 16(M) x 64(N) tile via 4
// v_wmma_f32_16x16x4_f32 accumulators.
// f32 A-frag layout: lane l (0-15) half h: a={A[m0+l][k0+2h], A[m0+l][k0+2h+1]}
// B-frag symmetric with N replacing M. C/D: vgpr j, lanes0-15: M=j; 16-31: M=j+8.
// ---------------------------------------------------------------------------
__global__ void __launch_bounds__(32)
wmma_gemm_f32(const float* __restrict__ A1, const float* __restrict__ W1,
              const float* __restrict__ b1,
              const float* __restrict__ A2, const float* __restrict__ W2,
              const float* __restrict__ b2,
              float* __restrict__ C, int K, int Nout, int act)
{
    const int lane = threadIdx.x & 31;
    const int hh   = lane >> 4;      // which 16-lane half
    const int l    = lane & 15;
    const int m0   = blockIdx.x * 16;
    const int n0   = blockIdx.y * 64;

    v8f acc[4];
    acc[0] = (v8f){}; acc[1] = (v8f){}; acc[2] = (v8f){}; acc[3] = (v8f){};

    gemm_accum(A1, W1, K, m0, n0, l, hh, acc);
    if (A2 != nullptr) gemm_accum(A2, W2, K, m0, n0, l, hh, acc);

    #pragma unroll
    for (int t = 0; t < 4; ++t) {
        const int n = n0 + t * 16 + l;
        float bias = b1[n] + (b2 ? b2[n] : 0.f);
        #pragma unroll
        for (int j = 0; j < 8; ++j) {
            const int row = m0 + j + 8 * hh;
            float v = acc[t][j] + bias;
            if (act == 1) v = fmaxf(v, 0.f);
            C[(size_t)row * Nout + n] = v;
        }
    }
}

// per-node attention dot products: s1 = x . attW[0:128], s2 = x . attW[128:256]
__global__ void node_att_dots(const float* __restrict__ x, const float* __restrict__ attW,
                              float* __restrict__ s1, float* __restrict__ s2)
{
    const int wid  = (int)((blockIdx.x * blockDim.x + threadIdx.x) >> 5);
    const int lane = threadIdx.x & 31;
    if (wid >= N_NODES) return;
    const float4 v  = ((const float4*)(x + (size_t)wid * HID))[lane];
    const float4 w1 = ((const float4*)attW)[lane];
    const float4 w2 = ((const float4*)(attW + HID))[lane];
    float p1 = v.x * w1.x + v.y * w1.y + v.z * w1.z + v.w * w1.w;
    float p2 = v.x * w2.x + v.y * w2.y + v.z * w2.z + v.w * w2.w;
    #pragma unroll
    for (int off = 16; off > 0; off >>= 1) {
        p1 += __shfl_xor(p1, off, 32);
        p2 += __shfl_xor(p2, off, 32);
    }
    if (lane == 0) { s1[wid] = p1; s2[wid] = p2; }
}

// one wave per edge: att = sigmoid(s1[src]+s2[dst]+b); agg[dst] += y[src]*att
__global__ void edge_gather_scatter(const int* __restrict__ src, const int* __restrict__ dst,
                                    const float* __restrict__ s1, const float* __restrict__ s2,
                                    const float* __restrict__ attb,
                                    const float* __restrict__ y, float* __restrict__ agg)
{
    const int wid  = (int)((blockIdx.x * blockDim.x + threadIdx.x) >> 5);
    const int lane = threadIdx.x & 31;
    if (wid >= N_EDGES) return;
    const int s = src[wid], d = dst[wid];
    const float att = 1.0f / (1.0f + expf(-(s1[s] + s2[d] + attb[0])));
    const float4 v = ((const float4*)(y + (size_t)s * HID))[lane];
    float* ar = agg + (size_t)d * HID + lane * 4;
    __hip_atomic_fetch_add(ar + 0, v.x * att, __ATOMIC_RELAXED, __HIP_MEMORY_SCOPE_AGENT);
    __hip_atomic_fetch_add(ar + 1, v.y * att, __ATOMIC_RELAXED, __HIP_MEMORY_SCOPE_AGENT);
    __hip_atomic_fetch_add(ar + 2, v.z * att, __ATOMIC_RELAXED, __HIP_MEMORY_SCOPE_AGENT);
    __hip_atomic_fetch_add(ar + 3, v.w * att, __ATOMIC_RELAXED, __HIP_MEMORY_SCOPE_AGENT);
}

// column-wise sum and sum-of-squares into sums[0:128] and sums[128:256]
__global__ void __launch_bounds__(128)
bn_stats(const float* __restrict__ in, float* __restrict__ sums)
{
    const int c  = threadIdx.x;           // 0..127
    const int r0 = blockIdx.x * 512;
    const int r1 = (r0 + 512 < N_NODES) ? (r0 + 512) : N_NODES;
    float s = 0.f, q = 0.f;
    for (int r = r0; r < r1; ++r) {
        const float v = in[(size_t)r * HID + c];
        s += v; q += v * v;
    }
    __hip_atomic_fetch_add(&sums[c],       s, __ATOMIC_RELAXED, __HIP_MEMORY_SCOPE_AGENT);
    __hip_atomic_fetch_add(&sums[HID + c], q, __ATOMIC_RELAXED, __HIP_MEMORY_SCOPE_AGENT);
}

__global__ void bn_apply(const float* __restrict__ in, float* __restrict__ out,
                         const float* __restrict__ sums,
                         const float* __restrict__ g, const float* __restrict__ beta, int n)
{
    const int i = blockIdx.x * blockDim.x + threadIdx.x;
    if (i >= n) return;
    const int c = i & (HID - 1);
    const float inv = 1.0f / (float)N_NODES;
    const float mu  = sums[c] * inv;
    const float var = sums[HID + c] * inv - mu * mu;
    out[i] = (in[i] - mu) * rsqrtf(var + EPSBN) * g[c] + beta[c];
}

__global__ void gru_gate(const float* __restrict__ R, const float* __restrict__ Z,
                         const float* __restrict__ T1, const float* __restrict__ T2,
                         const float* __restrict__ x, float* __restrict__ out, int n)
{
    const int i = blockIdx.x * blockDim.x + threadIdx.x;
    if (i >= n) return;
    const float r  = 1.0f / (1.0f + expf(-R[i]));
    const float z  = 1.0f / (1.0f + expf(-Z[i]));
    const float nn = tanhf(T1[i] + r * T2[i]);
    out[i] = (1.0f - z) * nn + z * x[i];
}

// sum-pool per graph: one wave per node, float4 atomics into grep[batch[node]]
__global__ void graph_scatter(const float* __restrict__ x, const int* __restrict__ batch,
                              float* __restrict__ grep)
{
    const int wid  = (int)((blockIdx.x * blockDim.x + threadIdx.x) >> 5);
    const int lane = threadIdx.x & 31;
    if (wid >= N_NODES) return;
    const int g = batch[wid];
    const float4 v = ((const float4*)(x + (size_t)wid * HID))[lane];
    float* gr = grep + (size_t)g * HID + lane * 4;
    __hip_atomic_fetch_add(gr + 0, v.x, __ATOMIC_RELAXED, __HIP_MEMORY_SCOPE_AGENT);
    __hip_atomic_fetch_add(gr + 1, v.y, __ATOMIC_RELAXED, __HIP_MEMORY_SCOPE_AGENT);
    __hip_atomic_fetch_add(gr + 2, v.z, __ATOMIC_RELAXED, __HIP_MEMORY_SCOPE_AGENT);
    __hip_atomic_fetch_add(gr + 3, v.w, __ATOMIC_RELAXED, __HIP_MEMORY_SCOPE_AGENT);
}

__global__ void readout_final(const float* __restrict__ h1, const float* __restrict__ W2,
                              const float* __restrict__ b2, float* __restrict__ out)
{
    const int g = threadIdx.x;
    if (g >= NUM_GRAPHS) return;
    float acc = b2[0];
    #pragma unroll
    for (int k = 0; k < 64; ++k) acc += h1[g * 64 + k] * W2[k];
    out[g] = acc;
}

// ---------------------------------------------------------------------------
extern "C" void kernel_launch(void* const* d_in, const int* in_sizes, int n_in,
                              void* d_out, int out_size, void* d_ws, size_t ws_size,
                              hipStream_t stream)
{
    (void)in_sizes; (void)n_in; (void)out_size; (void)ws_size;
    const float* node_f  = (const float*)d_in[0];
    const int*   eidx    = (const int*)d_in[1];
    const int*   src     = eidx;
    const int*   dstp    = eidx + N_EDGES;
    const int*   batch   = (const int*)d_in[2];
    const float* embW    = (const float*)d_in[3];
    const float* embB    = (const float*)d_in[4];
    const float* embG    = (const float*)d_in[5];
    const float* embBeta = (const float*)d_in[6];
    const float* msgW    = (const float*)d_in[7];
    const float* msgB    = (const float*)d_in[8];
    const float* attW    = (const float*)d_in[9];
    const float* attB    = (const float*)d_in[10];
    const float* Wih     = (const float*)d_in[11];
    const float* bih     = (const float*)d_in[12];
    const float* Whh     = (const float*)d_in[13];
    const float* bhh     = (const float*)d_in[14];
    const float* bnG     = (const float*)d_in[15];
    const float* bnB     = (const float*)d_in[16];
    const float* roW1    = (const float*)d_in[17];
    const float* roB1    = (const float*)d_in[18];
    const float* roW2    = (const float*)d_in[19];
    const float* roB2    = (const float*)d_in[20];
    float* out = (float*)d_out;

    const size_t NF = (size_t)N_NODES * HID;   // 6.4M floats
    float* ws   = (float*)d_ws;
    float* bufA = ws;            // current node features x
    float* bufB = bufA + NF;     // pre-BN h_new / pre-BN embedding
    float* ybuf = bufB + NF;     // x @ msg_W^T
    float* agg  = ybuf + NF;     // aggregated messages
    float* Rb   = agg  + NF;     // gate r pre-activation
    float* Zb   = Rb   + NF;     // gate z pre-activation
    float* T1b  = Zb   + NF;     // i_n
    float* T2b  = T1b  + NF;     // h_n
    float* s1   = T2b  + NF;
    float* s2   = s1 + N_NODES;
    float* bnS  = s2 + N_NODES;                 // 256 floats (sum, sumsq)
    float* grep = bnS + 2 * HID;                // 128x128
    float* h1   = grep + NUM_GRAPHS * HID;      // 128x64

    const dim3 gemmGridH(N_NODES / 16, HID / 64);   // 3125 x 2
    const int  nElem   = (int)NF;
    const int  ewBlks  = (nElem + 255) / 256;
    const int  waveN   = (N_NODES * 32 + 255) / 256;   // wave-per-node launches
    const int  waveE   = (N_EDGES * 32 + 255) / 256;   // wave-per-edge launches
    const int  bnBlks  = (N_NODES + 511) / 512;

    // ---- atom embedding: relu(node_f @ embW^T + embB), then BN ----
    wmma_gemm_f32<<<gemmGridH, 32, 0, stream>>>(node_f, embW, embB,
                                                nullptr, nullptr, nullptr,
                                                bufB, ATOM_DIM, HID, 1);
    hipMemsetAsync(bnS, 0, 2 * HID * sizeof(float), stream);
    bn_stats<<<bnBlks, 128, 0, stream>>>(bufB, bnS);
    bn_apply<<<ewBlks, 256, 0, stream>>>(bufB, bufA, bnS, embG, embBeta, nElem);

    // ---- message-passing layers ----
    for (int l = 0; l < N_LAYERS; ++l) {
        const float* mW  = msgW + (size_t)l * HID * HID;
        const float* mB  = msgB + (size_t)l * HID;
        const float* aW  = attW + (size_t)l * 2 * HID;
        const float* aB  = attB + l;
        const float* Wr  = Wih + (size_t)l * 3 * HID * HID;
        const float* Wz  = Wr + (size_t)HID * HID;
        const float* Wn  = Wz + (size_t)HID * HID;
        const float* br_ = bih + (size_t)l * 3 * HID;
        const float* bz_ = br_ + HID;
        const float* bn_ = bz_ + HID;
        const float* Ur  = Whh + (size_t)l * 3 * HID * HID;
        const float* Uz  = Ur + (size_t)HID * HID;
        const float* Un  = Uz + (size_t)HID * HID;
        const float* cr_ = bhh + (size_t)l * 3 * HID;
        const float* cz_ = cr_ + HID;
        const float* cn_ = cz_ + HID;

        // per-node attention dots + per-node message transform
        node_att_dots<<<waveN, 256, 0, stream>>>(bufA, aW, s1, s2);
        wmma_gemm_f32<<<gemmGridH, 32, 0, stream>>>(bufA, mW, mB,
                                                    nullptr, nullptr, nullptr,
                                                    ybuf, HID, HID, 0);
        // edge gather/scale/scatter into agg
        hipMemsetAsync(agg, 0, NF * sizeof(float), stream);
        edge_gather_scatter<<<waveE, 256, 0, stream>>>(src, dstp, s1, s2, aB, ybuf, agg);

        // GRU pre-activations (fused dual GEMMs for r and z)
        wmma_gemm_f32<<<gemmGridH, 32, 0, stream>>>(agg, Wr, br_, bufA, Ur, cr_, Rb,  HID, HID, 0);
        wmma_gemm_f32<<<gemmGridH, 32, 0, stream>>>(agg, Wz, bz_, bufA, Uz, cz_, Zb,  HID, HID, 0);
        wmma_gemm_f32<<<gemmGridH, 32, 0, stream>>>(agg, Wn, bn_, nullptr, nullptr, nullptr, T1b, HID, HID, 0);
        wmma_gemm_f32<<<gemmGridH, 32, 0, stream>>>(bufA, Un, cn_, nullptr, nullptr, nullptr, T2b, HID, HID, 0);
        gru_gate<<<ewBlks, 256, 0, stream>>>(Rb, Zb, T1b, T2b, bufA, bufB, nElem);

        // BN (training-mode, biased variance)
        hipMemsetAsync(bnS, 0, 2 * HID * sizeof(float), stream);
        bn_stats<<<bnBlks, 128, 0, stream>>>(bufB, bnS);
        bn_apply<<<ewBlks, 256, 0, stream>>>(bufB, bufA, bnS,
                                             bnG + (size_t)l * HID, bnB + (size_t)l * HID, nElem);
    }

    // ---- readout ----
    hipMemsetAsync(grep, 0, (size_t)NUM_GRAPHS * HID * sizeof(float), stream);
    graph_scatter<<<waveN, 256, 0, stream>>>(bufA, batch, grep);
    wmma_gemm_f32<<<dim3(NUM_GRAPHS / 16, 64 / 64), 32, 0, stream>>>(grep, roW1, roB1,
                                                                     nullptr, nullptr, nullptr,
                                                                     h1, HID, 64, 1);
    readout_final<<<1, 128, 0, stream>>>(h1, roW2, roB2, out);
}